// GCN_5007931867570
// MI455X (gfx1250) — compile-verified
//
#include <hip/hip_runtime.h>
#include <hip/hip_bf16.h>

typedef float v2f __attribute__((ext_vector_type(2)));
typedef float v8f __attribute__((ext_vector_type(8)));

#define D 128           // D_IN == D_OUT == 128
#define KSTEPS 32       // 128 / 4 (WMMA f32 K=4)
#define NTILES 8        // 128 / 16 output column tiles
#define TPB 256         // 8 waves per block
#define WAVES_PER_BLOCK 8

// ---------------------------------------------------------------------------
// Kernel 1: zero the workspace (summed[N*128] + deg[N])
// ---------------------------------------------------------------------------
__global__ void zero_ws(float* __restrict__ p, long long n) {
    long long i = (long long)blockIdx.x * blockDim.x + threadIdx.x;
    if (i < n) p[i] = 0.0f;
}

// ---------------------------------------------------------------------------
// Kernel 2: edge scatter-add. One wave32 per edge, float4 per lane.
//   summed[dst] += feature[src];  deg[dst] += 1
// Feature table + accumulators are L2-resident (25.6 MB each vs 192 MB L2),
// so the atomics execute at L2 throughput.
// ---------------------------------------------------------------------------
__global__ __launch_bounds__(TPB) void edge_scatter(
    const float* __restrict__ feature,
    const int*   __restrict__ src,
    const int*   __restrict__ dst,
    float*       __restrict__ summed,
    float*       __restrict__ deg,
    int E)
{
    long long gid  = (long long)blockIdx.x * blockDim.x + threadIdx.x;
    int       e    = (int)(gid >> 5);
    int       lane = (int)(gid & 31);
    if (e >= E) return;

    int s = src[e];
    int d = dst[e];

    const float4 f = ((const float4*)(feature + (long long)s * D))[lane];
    float* base = summed + (long long)d * D + lane * 4;
    atomicAdd(base + 0, f.x);
    atomicAdd(base + 1, f.y);
    atomicAdd(base + 2, f.z);
    atomicAdd(base + 3, f.w);
    if (lane == 0) atomicAdd(deg + d, 1.0f);
}

// ---------------------------------------------------------------------------
// Kernel 3: fused mean / zero-degree-fallback + Linear + ReLU via
// V_WMMA_F32_16X16X4_F32.  out[m][n] = relu( sum_k h[m][k]*W[n][k] + bias[n] )
//   A tile = h rows (16x4 per step), B tile = W^T (4x16 per step) from LDS.
// A-fragment lane layout (ISA 7.12.2, 32-bit A 16x4):
//   lanes 0-15:  row = lane,    holds K = {0,1} of the 4-wide K block
//   lanes 16-31: row = lane-16, holds K = {2,3}
// B-fragment (4x16, row striped across lanes within a VGPR):
//   lanes 0-15:  col = lane,    holds K = {0,1}  -> weight[col][k..k+1]
//   lanes 16-31: col = lane-16, holds K = {2,3}
// C/D layout: VGPR r -> M = r + 8*(lane>=16), N = lane&15.
// ---------------------------------------------------------------------------
__global__ __launch_bounds__(TPB) void gcn_wmma(
    const float* __restrict__ summed,
    const float* __restrict__ deg,
    const float* __restrict__ feature,
    const float* __restrict__ weight,   // [D_OUT][D_IN] row-major
    const float* __restrict__ bias,     // [D_OUT]
    float*       __restrict__ out,      // [N][D_OUT]
    int numTiles)
{
    __shared__ float wlds[D * D];       // 64 KB: full weight matrix

    // Cooperative stage of weight into LDS (float4 per thread-iter).
    {
        const float4* wg = (const float4*)weight;
        float4*       wl = (float4*)wlds;
        #pragma unroll
        for (int i = 0; i < (D * D / 4) / TPB; ++i)
            wl[threadIdx.x + i * TPB] = wg[threadIdx.x + i * TPB];
    }
    __syncthreads();

    const int wave = threadIdx.x >> 5;
    const int lane = threadIdx.x & 31;
    const int tile = blockIdx.x * WAVES_PER_BLOCK + wave;
    if (tile >= numTiles) return;       // after barrier: no divergent barrier

    const int row  = tile * 16 + (lane & 15);
    const int kh   = lane >> 4;         // 0 -> K{0,1}, 1 -> K{2,3}

    // Fused h = (deg>0) ? summed/deg : feature, built as A fragments in regs.
    const float dv    = deg[row];
    const bool  has   = dv > 0.0f;
    const float scale = has ? (1.0f / dv) : 1.0f;
    const float* arow = has ? (summed + (long long)row * D)
                            : (feature + (long long)row * D);

    v2f a[KSTEPS];
    #pragma unroll
    for (int kk = 0; kk < KSTEPS; ++kk) {
        v2f t = *(const v2f*)(arow + kk * 4 + kh * 2);   // global_load_b64
        t.x *= scale;
        t.y *= scale;
        a[kk] = t;
    }

    const float* wb = wlds + ((lane & 15) * D + kh * 2);

    #pragma unroll
    for (int nb = 0; nb < NTILES; ++nb) {
        v8f c = {};
        const float* wcol = wb + nb * 16 * D;
        #pragma unroll
        for (int kk = 0; kk < KSTEPS; ++kk) {
            const v2f b = *(const v2f*)(wcol + kk * 4);  // ds_load_b64
            c = __builtin_amdgcn_wmma_f32_16x16x4_f32(
                    /*neg_a=*/false, a[kk],
                    /*neg_b=*/false, b,
                    /*c_mod=*/(short)0, c,
                    /*reuse_a=*/false, /*reuse_b=*/false);
        }

        const int   col   = nb * 16 + (lane & 15);
        const float bv    = bias[col];
        float* op = out + ((long long)(tile * 16 + kh * 8)) * D + col;
        #pragma unroll
        for (int r = 0; r < 8; ++r) {
            float v = c[r] + bv;
            op[(long long)r * D] = v > 0.0f ? v : 0.0f;
        }
    }
}

// ---------------------------------------------------------------------------
extern "C" void kernel_launch(void* const* d_in, const int* in_sizes, int n_in,
                              void* d_out, int out_size, void* d_ws, size_t ws_size,
                              hipStream_t stream) {
    const float* feature = (const float*)d_in[0];
    const int*   src     = (const int*)d_in[1];
    const int*   dst     = (const int*)d_in[2];
    const float* weight  = (const float*)d_in[3];
    const float* bias    = (const float*)d_in[4];
    float*       out     = (float*)d_out;

    const int N = in_sizes[0] / D;      // 50000
    const int E = in_sizes[1];          // 1600000

    float* summed = (float*)d_ws;                        // N*128 floats
    float* deg    = summed + (long long)N * D;           // N floats

    // 1) zero accumulators
    long long ztotal = (long long)N * (D + 1);
    zero_ws<<<(unsigned)((ztotal + TPB - 1) / TPB), TPB, 0, stream>>>(summed, ztotal);

    // 2) edge scatter (wave per edge)
    long long sthreads = (long long)E * 32;
    edge_scatter<<<(unsigned)((sthreads + TPB - 1) / TPB), TPB, 0, stream>>>(
        feature, src, dst, summed, deg, E);

    // 3) fused mean + Linear + ReLU with f32 WMMA
    const int numTiles = N / 16;        // 3125 (N divisible by 16)
    const int grid     = (numTiles + WAVES_PER_BLOCK - 1) / WAVES_PER_BLOCK;
    gcn_wmma<<<grid, TPB, 0, stream>>>(summed, deg, feature, weight, bias, out, numTiles);
}